// MaskFormerCriterion_4097398800946
// MI455X (gfx1250) — compile-verified
//
#include <hip/hip_runtime.h>
#include <math.h>

// ---------------- problem constants (from reference) ----------------
#define NCLS   19          // NUM_CLASSES
#define C1     20          // NUM_CLASSES + 1 (incl. no-object)
#define BB     4
#define BQ     100
#define IH     512
#define IW     1024
#define MH     128
#define MW     256
#define HWPIX  (MH*MW)     // 32768
#define QT     7           // ceil(100/16) query tiles
#define SEG    16          // K segments per (b, qtile)
#define SEGLEN (HWPIX/SEG) // 2048
#define CHUNK  256         // pixels staged in LDS per inner pass
#define EPSF   1e-5f

typedef float v2f __attribute__((ext_vector_type(2)));
typedef float v8f __attribute__((ext_vector_type(8)));

__device__ __forceinline__ float softplusf(float x) {
    return (x > 20.f) ? x : log1pf(__expf(x));
}
// fast sigmoid: v_exp_f32 + v_add + v_rcp_f32 (no IEEE div-scale/div-fixup chains)
__device__ __forceinline__ float sigmoidf(float x) {
    return __builtin_amdgcn_rcpf(1.f + __expf(-x));
}

// ---------------- K1: nearest-downsample anno + per-class counts ----------------
__global__ __launch_bounds__(256)
void k_anno(const int* __restrict__ anno, unsigned char* __restrict__ anno8,
            int* __restrict__ gtSum) {
    const int b = blockIdx.x;
    __shared__ int hist[NCLS];
    for (int i = threadIdx.x; i < NCLS; i += blockDim.x) hist[i] = 0;
    __syncthreads();
    const int* ab = anno + (size_t)b * IH * IW;
    unsigned char* ob = anno8 + (size_t)b * HWPIX;
    for (int i = threadIdx.x; i < HWPIX; i += blockDim.x) {
        const int y = i >> 8, x = i & 255;           // MW = 256
        const int v = ab[(y * 4) * IW + x * 4];      // nearest: idx = i * (H/h)
        ob[i] = (unsigned char)v;
        if (v >= 0 && v < NCLS) atomicAdd(&hist[v], 1);
    }
    __syncthreads();
    for (int i = threadIdx.x; i < NCLS; i += blockDim.x)
        gtSum[b * NCLS + i] = hist[i];
}

// ---------------- K2: fused WMMA pass over pred_masks (read once) ----------------
// grid = BB*QT*SEG workgroups, 1 wave (32 threads) each.
// Per workgroup: 16 queries x 32 class columns (19 valid) over a 2048-pixel K segment.
// dotP = pred @ onehot.T, dotS = sigmoid(pred) @ onehot.T via V_WMMA_F32_16X16X4_F32.
// Sigmoid values are staged in LDS during the coalesced load pass so the WMMA loop
// is pure ds_load + cndmask + wmma.
__global__ __launch_bounds__(32)
void k_partials(const float* __restrict__ pred, const unsigned char* __restrict__ anno8,
                float* __restrict__ pP, float* __restrict__ pS,
                float* __restrict__ pSp, float* __restrict__ pSig) {
    const int blk  = blockIdx.x;
    const int seg  = blk % SEG;
    const int qt   = (blk / SEG) % QT;
    const int b    = blk / (SEG * QT);
    const int lane = threadIdx.x;

    __shared__ float predT[16][CHUNK];  // 16 KB raw pred tile
    __shared__ float sigT[16][CHUNK];   // 16 KB sigmoid(pred) tile
    __shared__ int   annoT[CHUNK];      // downsampled class ids for this chunk
    __shared__ float red[32][16];       // cross-lane reduction scratch

    v8f accP0 = {}; v8f accP1 = {};     // dotP, class tiles [0..15] / [16..31]
    v8f accS0 = {}; v8f accS1 = {};     // dotS
    float spAcc[16], sigAcc[16];
    #pragma unroll
    for (int r = 0; r < 16; ++r) { spAcc[r] = 0.f; sigAcc[r] = 0.f; }

    const int k0seg = seg * SEGLEN;
    const unsigned char* ab = anno8 + (size_t)b * HWPIX;

    for (int ch = 0; ch < SEGLEN; ch += CHUNK) {
        const int k0 = k0seg + ch;
        // stage anno chunk
        for (int i = lane; i < CHUNK; i += 32) annoT[i] = (int)ab[k0 + i];
        // stage pred + sigmoid tiles (coalesced) + fused per-query partial sums
        #pragma unroll 1
        for (int r = 0; r < 16; ++r) {
            const int q = qt * 16 + r;
            if (q < BQ) {
                const float* row = pred + ((size_t)(b * BQ + q)) * HWPIX + k0;
                for (int c = lane; c < CHUNK; c += 32) {
                    const float v = row[c];
                    const float s = sigmoidf(v);
                    predT[r][c] = v;
                    sigT[r][c]  = s;
                    spAcc[r]  += softplusf(v);
                    sigAcc[r] += s;
                }
            } else {
                for (int c = lane; c < CHUNK; c += 32) {
                    predT[r][c] = 0.f;
                    sigT[r][c]  = 0.f;
                }
            }
        }
        __syncthreads();

        // WMMA over the chunk, K step = 4 pixels
        const int row   = lane & 15;
        const int khalf = (lane >= 16) ? 2 : 0;   // ISA 16x4 A layout: upper lanes carry K=2,3
        const int cls0  = row;                    // B-tile 0: classes 0..15
        const int cls1  = 16 + row;               // B-tile 1: classes 16..31 (19..31 padding)
        for (int kk = 0; kk < CHUNK; kk += 4) {
            v2f aP; aP.x = predT[row][kk + khalf]; aP.y = predT[row][kk + khalf + 1];
            v2f aS; aS.x = sigT[row][kk + khalf];  aS.y = sigT[row][kk + khalf + 1];
            const int c0 = annoT[kk + khalf];
            const int c1 = annoT[kk + khalf + 1];
            v2f b0; b0.x = (c0 == cls0) ? 1.f : 0.f; b0.y = (c1 == cls0) ? 1.f : 0.f;
            v2f b1; b1.x = (c0 == cls1) ? 1.f : 0.f; b1.y = (c1 == cls1) ? 1.f : 0.f;
            accP0 = __builtin_amdgcn_wmma_f32_16x16x4_f32(false, aP, false, b0, (short)0, accP0, false, false);
            accP1 = __builtin_amdgcn_wmma_f32_16x16x4_f32(false, aP, false, b1, (short)0, accP1, false, false);
            accS0 = __builtin_amdgcn_wmma_f32_16x16x4_f32(false, aS, false, b0, (short)0, accS0, false, false);
            accS1 = __builtin_amdgcn_wmma_f32_16x16x4_f32(false, aS, false, b1, (short)0, accS1, false, false);
        }
        __syncthreads();
    }

    // write D partials. D layout: VGPR j, lanes 0-15 -> M=j, lanes 16-31 -> M=j+8; N=lane&15
    const size_t base = (size_t)blk * 16 * 32;
    const int n  = lane & 15;
    const int mh = (lane >= 16) ? 8 : 0;
    #pragma unroll
    for (int j = 0; j < 8; ++j) {
        const int m = j + mh;
        pP[base + m * 32 + n]      = accP0[j];
        pP[base + m * 32 + 16 + n] = accP1[j];
        pS[base + m * 32 + n]      = accS0[j];
        pS[base + m * 32 + 16 + n] = accS1[j];
    }
    // reduce softplus / sigmoid partials across the 32 lanes (fixed order)
    for (int r = 0; r < 16; ++r) red[lane][r] = spAcc[r];
    __syncthreads();
    if (lane < 16) {
        float s = 0.f;
        for (int l = 0; l < 32; ++l) s += red[l][lane];
        pSp[(size_t)blk * 16 + lane] = s;
    }
    __syncthreads();
    for (int r = 0; r < 16; ++r) red[lane][r] = sigAcc[r];
    __syncthreads();
    if (lane < 16) {
        float s = 0.f;
        for (int l = 0; l < 32; ++l) s += red[l][lane];
        pSig[(size_t)blk * 16 + lane] = s;
    }
}

// ---------------- K3: reduce segments, cost matrix, Hungarian, loss (per batch) ----------------
__global__ __launch_bounds__(256)
void k_match_loss(const float* __restrict__ logits, const int* __restrict__ gtSum,
                  const float* __restrict__ pP, const float* __restrict__ pS,
                  const float* __restrict__ pSp, const float* __restrict__ pSig,
                  float* __restrict__ lossPerB) {
    const int b = blockIdx.x;
    const int tid = threadIdx.x;

    __shared__ float dotP[BQ][32];
    __shared__ float dotS[BQ][32];
    __shared__ float spS[BQ], sigS[BQ], lseS[BQ];
    __shared__ double costT[NCLS][BQ];          // [M rows][Q cols] for Hungarian
    __shared__ int presentCls[NCLS];
    __shared__ int Mcnt;
    __shared__ int matchGt[BQ];                 // matched gt index per query, -1 = unmatched
    // Hungarian state (e-maxx potentials)
    __shared__ double hu[NCLS + 1], hv[BQ + 1], hminv[BQ + 1];
    __shared__ int hp[BQ + 1], hway[BQ + 1];
    __shared__ bool hused[BQ + 1];
    // per-query loss pieces
    __shared__ float perQ[BQ], nllU[BQ], spU[BQ];

    // 1) reduce segment partials (fixed order over SEG)
    for (int idx = tid; idx < BQ * 32; idx += 256) {
        const int q = idx >> 5, c = idx & 31;
        const int qt = q >> 4, m = q & 15;
        float sp_ = 0.f, ss_ = 0.f;
        for (int s = 0; s < SEG; ++s) {
            const size_t base = ((size_t)((b * QT + qt) * SEG + s)) * 512;
            sp_ += pP[base + m * 32 + c];
            ss_ += pS[base + m * 32 + c];
        }
        dotP[q][c] = sp_;
        dotS[q][c] = ss_;
    }
    for (int q = tid; q < BQ; q += 256) {
        const int qt = q >> 4, m = q & 15;
        float a = 0.f, c2 = 0.f;
        for (int s = 0; s < SEG; ++s) {
            const size_t base = ((size_t)((b * QT + qt) * SEG + s)) * 16;
            a  += pSp[base + m];
            c2 += pSig[base + m];
        }
        spS[q] = a; sigS[q] = c2;
        // log-sum-exp over the 20 class logits
        const float* lq = logits + ((size_t)(b * BQ + q)) * C1;
        float mx = lq[0];
        for (int c = 1; c < C1; ++c) mx = fmaxf(mx, lq[c]);
        float se = 0.f;
        for (int c = 0; c < C1; ++c) se += __expf(lq[c] - mx);
        lseS[q] = mx + __logf(se);
    }
    if (tid == 0) {
        int M = 0;
        for (int c = 0; c < NCLS; ++c)
            if (gtSum[b * NCLS + c] > 0) presentCls[M++] = c;
        Mcnt = M;
    }
    __syncthreads();
    const int M = Mcnt;

    // 2) cost matrix (match reference: class -softmax, bce matmul form, dice w/o eps-num)
    for (int idx = tid; idx < M * BQ; idx += 256) {
        const int mi = idx / BQ, q = idx % BQ;
        const int cls = presentCls[mi];
        const float* lq = logits + ((size_t)(b * BQ + q)) * C1;
        const float prob = __expf(lq[cls] - lseS[q]);
        const float classCost = -prob;
        const float bce = (spS[q] - dotP[q][cls]) / (float)HWPIX;
        const float gs = (float)gtSum[b * NCLS + cls];
        const float dice = 1.f - (2.f * dotS[q][cls]) / (sigS[q] + gs + EPSF);
        costT[mi][q] = (double)(1.0f * classCost + 20.0f * bce + 1.0f * dice);
    }
    __syncthreads();

    // 3) Hungarian on [M, BQ] (thread 0, serial by nature; M <= 19)
    if (tid == 0) {
        for (int q = 0; q < BQ; ++q) matchGt[q] = -1;
        if (M > 0) {
            const int n = M, mm = BQ;
            for (int i = 0; i <= n; ++i) hu[i] = 0.0;
            for (int j = 0; j <= mm; ++j) { hv[j] = 0.0; hp[j] = 0; hway[j] = 0; }
            for (int i = 1; i <= n; ++i) {
                hp[0] = i;
                int j0 = 0;
                for (int j = 0; j <= mm; ++j) { hminv[j] = 1e300; hused[j] = false; }
                while (true) {
                    hused[j0] = true;
                    const int i0 = hp[j0];
                    double delta = 1e300;
                    int j1 = 0;
                    for (int j = 1; j <= mm; ++j) {
                        if (!hused[j]) {
                            const double cur = costT[i0 - 1][j - 1] - hu[i0] - hv[j];
                            if (cur < hminv[j]) { hminv[j] = cur; hway[j] = j0; }
                            if (hminv[j] < delta) { delta = hminv[j]; j1 = j; }
                        }
                    }
                    for (int j = 0; j <= mm; ++j) {
                        if (hused[j]) { hu[hp[j]] += delta; hv[j] -= delta; }
                        else hminv[j] -= delta;
                    }
                    j0 = j1;
                    if (hp[j0] == 0) break;
                }
                while (j0) { const int j1 = hway[j0]; hp[j0] = hp[j1]; j0 = j1; }
            }
            for (int j = 1; j <= mm; ++j)
                if (hp[j] != 0) matchGt[j - 1] = hp[j] - 1;
        }
    }
    __syncthreads();

    // 4) per-query loss pieces
    for (int q = tid; q < BQ; q += 256) {
        const float* lq = logits + ((size_t)(b * BQ + q)) * C1;
        const float noobjNll = -(lq[NCLS] - lseS[q]);
        const int mi = matchGt[q];
        if (mi >= 0) {
            const int cls = presentCls[mi];
            const float ce = -(lq[cls] - lseS[q]);
            const float bce = (spS[q] - dotP[q][cls]) / (float)HWPIX;
            const float gs = (float)gtSum[b * NCLS + cls];
            const float dice = 1.f - (2.f * dotS[q][cls] + EPSF) / (sigS[q] + gs + EPSF);
            perQ[q] = 2.0f * ce + 5.0f * bce + 5.0f * dice;
            nllU[q] = 0.f; spU[q] = 0.f;
        } else {
            perQ[q] = 0.f;
            nllU[q] = noobjNll;
            spU[q]  = spS[q];
        }
    }
    __syncthreads();
    if (tid == 0) {
        float lb = 0.f;
        if (M == 0) {
            float s = 0.f;
            for (int q = 0; q < BQ; ++q) s += nllU[q];
            lb = 0.1f * s / (float)BQ;
        } else {
            float s = 0.f, nl = 0.f, su = 0.f;
            int U = 0;
            for (int q = 0; q < BQ; ++q) {
                s += perQ[q];
                if (matchGt[q] < 0) { nl += nllU[q]; su += spU[q]; ++U; }
            }
            lb = s;
            if (U > 0)
                lb += 0.1f * (nl / (float)U) + 0.1f * (su / ((float)U * (float)HWPIX));
        }
        lossPerB[b] = lb;
    }
}

// ---------------- K4: final average ----------------
__global__ void k_final(const float* __restrict__ lossPerB, float* __restrict__ out) {
    float s = 0.f;
    for (int b = 0; b < BB; ++b) s += lossPerB[b];
    out[0] = s / (float)BB;
}

// ---------------- host-side launch ----------------
extern "C" void kernel_launch(void* const* d_in, const int* in_sizes, int n_in,
                              void* d_out, int out_size, void* d_ws, size_t ws_size,
                              hipStream_t stream) {
    const float* class_logits = (const float*)d_in[0];   // [4,100,20]
    const float* pred_masks   = (const float*)d_in[1];   // [4,100,128,256]
    const int*   anno         = (const int*)d_in[2];     // [4,512,1024]
    float* out = (float*)d_out;

    // workspace carve-up (16B aligned sections)
    char* ws = (char*)d_ws;
    unsigned char* anno8 = (unsigned char*)ws;                 ws += (size_t)BB * HWPIX;            // 128 KB
    int*   gtSum  = (int*)ws;                                  ws += 512;                           // 4*19 ints
    float* pP     = (float*)ws;                                ws += (size_t)BB*QT*SEG*16*32*4;     // 896 KB
    float* pS     = (float*)ws;                                ws += (size_t)BB*QT*SEG*16*32*4;     // 896 KB
    float* pSp    = (float*)ws;                                ws += (size_t)BB*QT*SEG*16*4;        // 28 KB
    float* pSig   = (float*)ws;                                ws += (size_t)BB*QT*SEG*16*4;        // 28 KB
    float* lossPerB = (float*)ws;                              ws += 64;

    k_anno<<<BB, 256, 0, stream>>>(anno, anno8, gtSum);
    k_partials<<<BB * QT * SEG, 32, 0, stream>>>(pred_masks, anno8, pP, pS, pSp, pSig);
    k_match_loss<<<BB, 256, 0, stream>>>(class_logits, gtSum, pP, pS, pSp, pSig, lossPerB);
    k_final<<<1, 1, 0, stream>>>(lossPerB, out);
}